// GCNLayer_68779606278427
// MI455X (gfx1250) — compile-verified
//
#include <hip/hip_runtime.h>

typedef __attribute__((ext_vector_type(2))) float v2f;
typedef __attribute__((ext_vector_type(8))) float v8f;

#define N_NODES 100000
#define N_EDGES 1600000
#define D_IN    128
#define D_OUT   64
#define M_TILES (N_NODES / 16)   // 6250, exact
#define KSTEPS  (D_IN / 4)       // 32 k-steps of K=4
#define NTILES  (D_OUT / 16)     // 4 n-tiles

// ---------------------------------------------------------------------------
// Phase 1: support = X @ W  using V_WMMA_F32_16X16X4_F32 (full fp32 precision)
// One wave computes a 16x64 tile (4 WMMA accumulators).
// W is staged in LDS PRE-SWIZZLED into B-fragment layout so the inner loop is
// one aligned ds_load_b64 per fragment (no register packing moves):
//   fragment pair for (s, j, lane) = { W[k][n], W[k+1][n] }
//   with k = 4s + 2*(lane>=16), n = 16j + (lane&15)
// ---------------------------------------------------------------------------
__global__ __launch_bounds__(256) void gcn_gemm(const float* __restrict__ X,
                                                const float* __restrict__ W,
                                                float* __restrict__ support) {
    __shared__ v2f Wfrag[KSTEPS * NTILES * 32];   // 32 KB of the 320 KB WGP LDS

    // Cooperative swizzled staging: 4096 fragment pairs / 256 threads = 16 each
    for (int idx = threadIdx.x; idx < KSTEPS * NTILES * 32; idx += 256) {
        const int s    = idx >> 7;          // k-step
        const int j    = (idx >> 5) & 3;    // n-tile
        const int ln   = idx & 31;          // lane the fragment belongs to
        const int half = ln >> 4;
        const int l16  = ln & 15;
        const int k    = 4 * s + 2 * half;
        const int n    = 16 * j + l16;
        v2f p;
        p.x = W[(size_t)k * D_OUT + n];
        p.y = W[(size_t)(k + 1) * D_OUT + n];
        Wfrag[idx] = p;
    }
    __syncthreads();

    const int wave   = threadIdx.x >> 5;          // 8 waves per block (wave32)
    const int lane   = threadIdx.x & 31;
    const int tile_m = blockIdx.x * 8 + wave;
    if (tile_m >= M_TILES) return;                // uniform per-wave exit: EXEC all-1s

    const int half = lane >> 4;                   // 0: lanes 0-15, 1: lanes 16-31
    const int l16  = lane & 15;
    const int row  = tile_m * 16 + l16;           // A-matrix row held by this lane

    v8f acc0 = {}, acc1 = {}, acc2 = {}, acc3 = {};
    const float* xrow = X + (size_t)row * D_IN + 2 * half;
    const v2f*   bp   = &Wfrag[lane];

    for (int s = 0; s < KSTEPS; ++s) {
        // A fragment (16x4 f32): half 0 holds K=4s,4s+1; half 1 holds K=4s+2,4s+3
        v2f a = *(const v2f*)(xrow + 4 * s);

        // B fragments: single aligned ds_load_b64 each, offsets 0/32/64/96 pairs
        const v2f* b = bp + s * (NTILES * 32);
        v2f b0 = b[0];
        v2f b1 = b[32];
        v2f b2 = b[64];
        v2f b3 = b[96];

        acc0 = __builtin_amdgcn_wmma_f32_16x16x4_f32(false, a, false, b0, (short)0, acc0, false, false);
        acc1 = __builtin_amdgcn_wmma_f32_16x16x4_f32(false, a, false, b1, (short)0, acc1, false, false);
        acc2 = __builtin_amdgcn_wmma_f32_16x16x4_f32(false, a, false, b2, (short)0, acc2, false, false);
        acc3 = __builtin_amdgcn_wmma_f32_16x16x4_f32(false, a, false, b3, (short)0, acc3, false, false);
    }

    // C/D layout: VGPR r, lanes 0-15 -> M=r, lanes 16-31 -> M=r+8; N = n0 + l16
    float* sp = support + (size_t)(tile_m * 16) * D_OUT;
    #pragma unroll
    for (int r = 0; r < 8; ++r) {
        const int m = r + 8 * half;
        sp[(size_t)m * D_OUT +  0 + l16] = acc0[r];
        sp[(size_t)m * D_OUT + 16 + l16] = acc1[r];
        sp[(size_t)m * D_OUT + 32 + l16] = acc2[r];
        sp[(size_t)m * D_OUT + 48 + l16] = acc3[r];
    }
}

// ---------------------------------------------------------------------------
// Phase 2: out[n][c] = bias[c]  (bias folded into init; atomics add on top)
// ---------------------------------------------------------------------------
__global__ __launch_bounds__(256) void gcn_init(float* __restrict__ out,
                                                const float* __restrict__ bias) {
    const size_t i = (size_t)blockIdx.x * blockDim.x + threadIdx.x;   // one float4 each
    if (i < ((size_t)N_NODES * D_OUT) / 4) {
        ((float4*)out)[i] = ((const float4*)bias)[i & (D_OUT / 4 - 1)];
    }
}

// ---------------------------------------------------------------------------
// Phase 3: scatter-add  out[row] += support[col] * val   (16 lanes per edge)
// ---------------------------------------------------------------------------
__global__ __launch_bounds__(256) void gcn_edges(const float* __restrict__ support,
                                                 const int* __restrict__ erow,
                                                 const int* __restrict__ ecol,
                                                 const float* __restrict__ eval,
                                                 float* __restrict__ out) {
    const long long t = (long long)blockIdx.x * blockDim.x + threadIdx.x;
    const int e = (int)(t >> 4);
    const int l = (int)(t & 15);
    if (e >= N_EDGES) return;

    const int   src = ecol[e];
    const int   dst = erow[e];
    const float v   = eval[e];

    const float4 s = *(const float4*)(support + (size_t)src * D_OUT + l * 4);
    float* o = out + (size_t)dst * D_OUT + l * 4;
    atomicAdd(o + 0, s.x * v);
    atomicAdd(o + 1, s.y * v);
    atomicAdd(o + 2, s.z * v);
    atomicAdd(o + 3, s.w * v);
}

// ---------------------------------------------------------------------------
extern "C" void kernel_launch(void* const* d_in, const int* in_sizes, int n_in,
                              void* d_out, int out_size, void* d_ws, size_t ws_size,
                              hipStream_t stream) {
    const float* x        = (const float*)d_in[0];
    const int*   edge_row = (const int*)  d_in[1];
    const int*   edge_col = (const int*)  d_in[2];
    const float* edge_val = (const float*)d_in[3];
    const float* weight   = (const float*)d_in[4];
    const float* bias     = (const float*)d_in[5];
    float*       out      = (float*)d_out;
    float*       support  = (float*)d_ws;   // 100000*64*4 = 25.6 MB scratch

    // Phase 1: dense GEMM (WMMA). 8 waves/block -> 8 row tiles per block.
    {
        const int blocks = (M_TILES + 7) / 8;   // 782
        gcn_gemm<<<blocks, 256, 0, stream>>>(x, weight, support);
    }
    // Phase 2: init output with bias.
    {
        const size_t n4 = ((size_t)N_NODES * D_OUT) / 4;  // 1.6M float4
        const int blocks = (int)((n4 + 255) / 256);
        gcn_init<<<blocks, 256, 0, stream>>>(out, bias);
    }
    // Phase 3: edge gather / scale / scatter-add.
    {
        const long long threads = (long long)N_EDGES * 16;  // 25.6M
        const int blocks = (int)((threads + 255) / 256);
        gcn_edges<<<blocks, 256, 0, stream>>>(support, edge_row, edge_col, edge_val, out);
    }
}